// FIRApply_19490561590062
// MI455X (gfx1250) — compile-verified
//
#include <hip/hip_runtime.h>
#include <hip/hip_bf16.h>

// Causal 16-tap FIR as WMMA f32 Toeplitz matmul (MI455X / gfx1250).
//
//   y[t0+16m+n] = D[m][n],  D = A x B,  K = 32 (8 x v_wmma_f32_16x16x4_f32)
//   A[m][k] = x[t0 + 16m + k - 16]                  (contiguous, 8B-aligned b64 loads)
//   B[k][n] = (0 <= n+16-k < 16) ? b[n+16-k] : 0    (constant banded tap matrix, built once)
//
// Memory-bound: ~246 MB total traffic -> ~10.6 us floor at 23.3 TB/s. x (122.9 MB)
// fits in the 192 MB L2, so the ~1.9x overlapped A-fragment reads stay on-chip;
// y stores use the non-temporal hint so the write stream does not evict x.
// The t0==0 zero-padding tile is peeled; it uses clamp+select (unconditional
// loads from max(i0,0) + v_cndmask to zero) so no exec-mask juggling anywhere.

typedef __attribute__((ext_vector_type(2))) float v2f;
typedef __attribute__((ext_vector_type(8))) float v8f;

namespace {
constexpr int kT           = 480000;      // samples per batch row
constexpr int kTaps        = 16;
constexpr int kTile        = 256;         // outputs per wave-tile (16x16 D)
constexpr int kTilesPerRow = kT / kTile;  // 1875 (exact)
}

__global__ __launch_bounds__(256) void fir_wmma_f32_kernel(
    const float* __restrict__ x,
    const float* __restrict__ taps,
    float* __restrict__ y)
{
    const int lane        = threadIdx.x & 31;   // wave32
    const int waveInBlock = threadIdx.x >> 5;   // 0..7
    const int row         = blockIdx.y;         // batch row, 0..63
    const int wavesPerRow = gridDim.x * (blockDim.x >> 5);
    const int waveId      = blockIdx.x * (blockDim.x >> 5) + waveInBlock;

    const float* __restrict__ xr = x + (size_t)row * kT;
    float* __restrict__       yr = y + (size_t)row * kT;

    const int n   = lane & 15;   // B column / D column / A row (same lane map)
    const int hi  = lane >> 4;   // which half of the wave
    const int klo = hi * 2;      // k offset within a K=4 chunk: frag.x -> klo, frag.y -> klo+1

    // ---- Build constant B fragments once per wave ----
    // chunk c, frag component v: global k = 4c + klo + v; tap index j = n + 16 - k.
    v2f bfrag[8];
#pragma unroll
    for (int c = 0; c < 8; ++c) {
        const int k0 = 4 * c + klo;
        const int j0 = n + 16 - k0;        // for frag.x
        const int j1 = j0 - 1;             // for frag.y
        bfrag[c].x = (j0 >= 0 && j0 < kTaps) ? taps[j0] : 0.0f;
        bfrag[c].y = (j1 >= 0 && j1 < kTaps) ? taps[j1] : 0.0f;
    }

    const int m = n;  // A-matrix row held by this lane

    int tile = waveId;

    // ---- Peeled first tile (t0 == 0): zero-pad reads below t=0 ----
    // Branch is wave-uniform (waveId is uniform per wave), so EXEC stays all-1s
    // for the WMMAs. Loads are unconditional (clamped address), results selected.
    if (tile == 0) {
        const int base = 16 * m + klo - 16;  // >= -16
        v8f acc = {};
#pragma unroll
        for (int c = 0; c < 8; ++c) {
            const int i0 = base + 4 * c;
            const int i1 = i0 + 1;
            const float v0 = xr[i0 < 0 ? 0 : i0];  // always valid, no branch
            const float v1 = xr[i1 < 0 ? 0 : i1];
            v2f a;
            a.x = (i0 < 0) ? 0.0f : v0;            // v_cndmask
            a.y = (i1 < 0) ? 0.0f : v1;
            acc = __builtin_amdgcn_wmma_f32_16x16x4_f32(
                false, a, false, bfrag[c], (short)0, acc, false, false);
        }
#pragma unroll
        for (int v = 0; v < 8; ++v) {
            __builtin_nontemporal_store(acc[v], &yr[16 * (v + 8 * hi) + n]);
        }
        tile += wavesPerRow;
    }

    // ---- Steady state: branch-free clause'd b64 loads -> 8x WMMA -> NT stores ----
    for (; tile < kTilesPerRow; tile += wavesPerRow) {
        const int t0   = tile * kTile;
        const int base = t0 + 16 * m + klo - 16;  // >= 240 here; 8B aligned (even index)

        v8f acc = {};
#pragma unroll
        for (int c = 0; c < 8; ++c) {
            const v2f a = *(const v2f*)(xr + base + 4 * c);
            acc = __builtin_amdgcn_wmma_f32_16x16x4_f32(
                false, a, false, bfrag[c], (short)0, acc, false, false);
        }

        // Lane holds D[v + 8*hi][n] in acc[v]; tile store covers a contiguous 1 KB.
        float* __restrict__ yt = yr + t0;
#pragma unroll
        for (int v = 0; v < 8; ++v) {
            __builtin_nontemporal_store(acc[v], &yt[16 * (v + 8 * hi) + n]);
        }
    }
}

extern "C" void kernel_launch(void* const* d_in, const int* in_sizes, int n_in,
                              void* d_out, int out_size, void* d_ws, size_t ws_size,
                              hipStream_t stream) {
    const float* x    = (const float*)d_in[0];  // [64, 480000] fp32
    const float* taps = (const float*)d_in[1];  // [16] fp32
    float*       y    = (float*)d_out;          // [64, 480000] fp32

    // 40 blocks x 8 waves = 320 waves per batch row (~6 tiles each),
    // 64 rows in grid.y -> 20480 waves total.
    dim3 grid(40, 64);
    fir_wmma_f32_kernel<<<grid, 256, 0, stream>>>(x, taps, y);
}